// TaMM_28965259444337
// MI455X (gfx1250) — compile-verified
//
#include <hip/hip_runtime.h>
#include <math.h>

typedef __attribute__((ext_vector_type(2))) float v2f;
typedef __attribute__((ext_vector_type(8))) float v8f;

#define HIDDEN 128
#define LKDIM  512
#define LQDIM  512
#define NBATCH 4
#define INV_TEMPER 0.08838834764831845f   // 1/sqrt(128)

// One block per (batch, 16-row q tile). 256 threads = 8 wave32.
__launch_bounds__(256)
__global__ void tamm_attn_kernel(const float* __restrict__ hidden,
                                 const int*   __restrict__ key_seq,
                                 const int*   __restrict__ value_matrix,
                                 const int*   __restrict__ key_mask,
                                 const float* __restrict__ key_table,
                                 const float* __restrict__ val_table,
                                 float*       __restrict__ out) {
  __shared__ float sP[16][LKDIM];     // exp(u)*mask, unnormalized (32 KB)
  __shared__ float sOut[16][HIDDEN];  // ko * invDenom (8 KB)
  __shared__ int   sKid[LKDIM];       // staged key ids (2 KB)
  __shared__ float sSum[16];
  __shared__ float sInv[16];

  const int tid  = threadIdx.x;
  const int wave = tid >> 5;
  const int lane = tid & 31;
  const int m    = lane & 15;          // WMMA row/col index within fragment
  const int kb   = (lane >> 4) << 1;   // K-subposition: 0 or 2
  const int mrb  = (lane >> 4) << 3;   // D-row base: 0 or 8

  const int b  = blockIdx.x >> 5;      // 512/16 = 32 q tiles per batch
  const int qt = blockIdx.x & 31;
  const int q0 = qt << 4;

  // Stage key ids for this batch; clear row sums.
  for (int i = tid; i < LKDIM; i += 256) sKid[i] = key_seq[b * LKDIM + i];
  if (tid < 16) sSum[tid] = 0.0f;
  __syncthreads();

  // Resident A fragments for the whole H=128 reduction (32 chunks of K=4).
  // 16x4 f32 A layout: lanes 0-15 hold M=lane, VGPR v holds K = 4c + kb + v.
  const float* hrow = hidden + (size_t)(b * LQDIM + q0 + m) * HIDDEN;
  v2f aF[32];
  #pragma unroll
  for (int c = 0; c < 32; ++c) {
    aF[c].x = hrow[c * 4 + kb];
    aF[c].y = hrow[c * 4 + kb + 1];
  }

  // ---- Phase 1: u = Q * EK^T (WMMA f32 16x16x4), masked exp into sP ----
  for (int t = wave; t < 32; t += 8) {
    const int kg = t * 16 + m;                         // score column / key index
    const float* krow = key_table + (size_t)sKid[kg] * HIDDEN;
    v8f acc = {};
    #pragma unroll
    for (int c = 0; c < 32; ++c) {
      v2f bF;                                          // B(h, n): h = 4c+kb+v, n = m
      bF.x = krow[c * 4 + kb];
      bF.y = krow[c * 4 + kb + 1];
      acc = __builtin_amdgcn_wmma_f32_16x16x4_f32(false, aF[c], false, bF,
                                                  (short)0, acc, false, false);
    }
    #pragma unroll
    for (int r = 0; r < 8; ++r) {
      const int mr = r + mrb;                          // D row = q row
      const float u = acc[r] * INV_TEMPER;
      const int msk = key_mask[(size_t)(b * LQDIM + q0 + mr) * LKDIM + kg];
      const float e = (msk > 0) ? expf(u) : 0.0f;
      sP[mr][kg] = e;
      float s = e;                                     // reduce 16 lanes (one row)
      s += __shfl_xor(s, 1);
      s += __shfl_xor(s, 2);
      s += __shfl_xor(s, 4);
      s += __shfl_xor(s, 8);
      if (m == 0) atomicAdd(&sSum[mr], s);
    }
  }
  __syncthreads();
  if (tid < 16) sInv[tid] = 1.0f / (sSum[tid] + 1e-10f);
  __syncthreads();

  // ---- Phase 2: ko = P * EK (WMMA), one 16-wide h tile per wave ----
  {
    const int h = wave * 16 + m;                       // output h column
    v8f acc = {};
    for (int c = 0; c < 128; ++c) {                    // K = 512 in chunks of 4
      const int k0 = c * 4 + kb;
      v2f aP, bK;
      aP.x = sP[m][k0];                                // A(m, k)
      aP.y = sP[m][k0 + 1];
      bK.x = key_table[(size_t)sKid[k0]     * HIDDEN + h];  // B(k, h)
      bK.y = key_table[(size_t)sKid[k0 + 1] * HIDDEN + h];
      acc = __builtin_amdgcn_wmma_f32_16x16x4_f32(false, aP, false, bK,
                                                  (short)0, acc, false, false);
    }
    #pragma unroll
    for (int r = 0; r < 8; ++r) {
      const int mr = r + mrb;
      sOut[mr][h] = acc[r] * sInv[mr];
    }
  }
  __syncthreads();

  // ---- Phase 3: o = sum_k p * val_table[vm[q,k]]; fuse o+ko+hidden ----
  const int h0 = lane * 4;
  #pragma unroll
  for (int rq = 0; rq < 2; ++rq) {
    const int qr = wave * 2 + rq;                      // 8 waves x 2 rows = 16
    const size_t rowbase = (size_t)(b * LQDIM + q0 + qr);
    const int* vmrow = value_matrix + rowbase * LKDIM;
    float4 acc = make_float4(0.f, 0.f, 0.f, 0.f);
    for (int k = 0; k < LKDIM; ++k) {
      const float w = sP[qr][k];                       // uniform across lanes
      if (w != 0.0f) {                                 // skip masked gathers
        if (k + 16 < LKDIM)
          __builtin_prefetch(val_table + (size_t)vmrow[k + 16] * HIDDEN + h0, 0, 1);
        const float4 v = *(const float4*)(val_table + (size_t)vmrow[k] * HIDDEN + h0);
        acc.x += w * v.x; acc.y += w * v.y;
        acc.z += w * v.z; acc.w += w * v.w;
      }
    }
    const float inv = sInv[qr];
    const float4 hv = *(const float4*)(hidden + rowbase * HIDDEN + h0);
    float4 res;
    res.x = acc.x * inv + sOut[qr][h0 + 0] + hv.x;
    res.y = acc.y * inv + sOut[qr][h0 + 1] + hv.y;
    res.z = acc.z * inv + sOut[qr][h0 + 2] + hv.z;
    res.w = acc.w * inv + sOut[qr][h0 + 3] + hv.w;
    *(float4*)(out + rowbase * HIDDEN + h0) = res;
  }
}

extern "C" void kernel_launch(void* const* d_in, const int* in_sizes, int n_in,
                              void* d_out, int out_size, void* d_ws, size_t ws_size,
                              hipStream_t stream) {
  const float* hidden       = (const float*)d_in[0];
  const int*   key_seq      = (const int*)d_in[1];
  const int*   value_matrix = (const int*)d_in[2];
  const int*   key_mask     = (const int*)d_in[3];
  const float* key_table    = (const float*)d_in[4];
  const float* val_table    = (const float*)d_in[5];
  float* out = (float*)d_out;

  dim3 grid(NBATCH * (LQDIM / 16));   // 128 blocks
  dim3 block(256);                    // 8 wave32
  hipLaunchKernelGGL(tamm_attn_kernel, grid, block, 0, stream,
                     hidden, key_seq, value_matrix, key_mask,
                     key_table, val_table, out);
}